// ProjectLayer_32366873543427
// MI455X (gfx1250) — compile-verified
//
#include <hip/hip_runtime.h>

typedef __attribute__((ext_vector_type(2))) float v2f;
typedef __attribute__((ext_vector_type(8))) float v8f;

#define HM_W 240
#define HM_H 128
#define NCAM 5
#define NJ   15
#define NVOX 128000   // 80*80*20
#define NB   8

__device__ __forceinline__ void voxel_coords(int v, float cx, float cy, float cz,
                                             float& x, float& y, float& z) {
  // meshgrid 'ij', ravel: v = (vx*80 + vy)*20 + vz
  int vz = v % 20;
  int r  = v / 20;
  int vy = r % 80;
  int vx = r / 80;
  x = fmaf((float)vx, 8000.0f / 79.0f, -4000.0f) + cx;
  y = fmaf((float)vy, 8000.0f / 79.0f, -4000.0f) + cy;
  z = fmaf((float)vz, 2000.0f / 19.0f, -1000.0f) + cz;
}

__global__ __launch_bounds__(256) void project_layer_kernel(
    const float* __restrict__ heat,     // [B,5,15,128,240]
    const float* __restrict__ gcenter,  // [B,3]
    const float* __restrict__ Rm,       // [B,5,3,3]
    const float* __restrict__ Tm,       // [B,5,3]
    const float* __restrict__ fo,       // [B,5,2]
    const float* __restrict__ co,       // [B,5,2]
    const float* __restrict__ ko,       // [B,5,3]
    const float* __restrict__ po,       // [B,5,2]
    const float* __restrict__ cco,      // [B,5,2]
    const float* __restrict__ sco,      // [B,5,2]
    float* __restrict__ cubes,          // [B,15,128000]
    float* __restrict__ grids)          // [B,128000,3]
{
  const int b    = blockIdx.y;
  const int tid  = threadIdx.x;
  const int lane = tid & 31;
  const int l16  = lane & 15;
  const bool hi  = lane >= 16;
  const int tileBase = blockIdx.x * 256 + (tid >> 5) * 32;  // wave's 32-voxel tile
  const int v = tileBase + lane;                            // this thread's voxel

  const float cx = gcenter[b * 3 + 0];
  const float cy = gcenter[b * 3 + 1];
  const float cz = gcenter[b * 3 + 2];

  // ---------------- A matrix: 16 rows = (cam,axis) of [ R_row | -R.T ] -----
  // 32-bit 16x4 A layout: lanes 0-15 hold K=0,1 ; lanes 16-31 hold K=2,3 of row (lane&15)
  v2f a; a.x = 0.0f; a.y = 0.0f;
  if (l16 < 15) {
    int cam  = l16 / 3;
    int axis = l16 - 3 * cam;
    const float* Rr = Rm + (((size_t)(b * NCAM + cam) * 3 + axis) * 3);
    const float* Tt = Tm + ((size_t)(b * NCAM + cam) * 3);
    if (!hi) { a.x = Rr[0]; a.y = Rr[1]; }
    else {
      a.x = Rr[2];
      a.y = -(Rr[0] * Tt[0] + Rr[1] * Tt[1] + Rr[2] * Tt[2]);
    }
  }

  // ---------------- B matrices: columns = voxel [x,y,z,1] ------------------
  // 4x16 B layout: lanes 0-15 hold rows K=0,1 ; lanes 16-31 hold rows K=2,3 for col (lane&15)
  float x1, y1, z1, x2, y2, z2;
  voxel_coords(tileBase + l16,      cx, cy, cz, x1, y1, z1);
  voxel_coords(tileBase + 16 + l16, cx, cy, cz, x2, y2, z2);
  v2f b1, b2;
  if (!hi) { b1.x = x1; b1.y = y1; b2.x = x2; b2.y = y2; }
  else     { b1.x = z1; b1.y = 1.0f; b2.x = z2; b2.y = 1.0f; }

  v8f czero = {0.f, 0.f, 0.f, 0.f, 0.f, 0.f, 0.f, 0.f};
  // D[m][n] = R.p - R.T = xcam row m (m = 3*cam+axis) for voxel n
  v8f d1 = __builtin_amdgcn_wmma_f32_16x16x4_f32(false, a, false, b1, (short)0, czero, false, false);
  v8f d2 = __builtin_amdgcn_wmma_f32_16x16x4_f32(false, a, false, b2, (short)0, czero, false, false);

  // Redistribute: voxel n's rows 0-7 live in lane n, rows 8-15 in lane n+16.
  float xc[16];
#pragma unroll
  for (int m = 0; m < 8; ++m) {
    float o1 = __shfl(d1[m], lane ^ 16, 32);
    float o2 = __shfl(d2[m], lane ^ 16, 32);
    if (!hi) { xc[m] = d1[m]; xc[8 + m] = o1; }
    else     { xc[m] = o2;    xc[8 + m] = d2[m]; }
  }

  const float gx = hi ? x2 : x1;
  const float gy = hi ? y2 : y1;
  const float gz = hi ? z2 : z1;

  float num[NJ];
#pragma unroll
  for (int j = 0; j < NJ; ++j) num[j] = 0.0f;
  float den = 1e-6f;

  for (int n = 0; n < NCAM; ++n) {
    const int bn = b * NCAM + n;   // wave-uniform -> scalar loads
    const float f0  = fo[bn * 2 + 0],  f1  = fo[bn * 2 + 1];
    const float c0  = co[bn * 2 + 0],  c1  = co[bn * 2 + 1];
    const float k0  = ko[bn * 3 + 0],  k1  = ko[bn * 3 + 1], k2 = ko[bn * 3 + 2];
    const float p0  = po[bn * 2 + 0],  p1  = po[bn * 2 + 1];
    const float cc0 = cco[bn * 2 + 0], cc1 = cco[bn * 2 + 1];
    const float sc0 = sco[bn * 2 + 0];

    const float xcx = xc[3 * n + 0];
    const float xcy = xc[3 * n + 1];
    const float xcz = xc[3 * n + 2];

    const float inv = 1.0f / (xcz + 1e-5f);
    const float yn0 = xcx * inv, yn1 = xcy * inv;
    const float r2 = yn0 * yn0 + yn1 * yn1;
    const float radial = 1.0f + k0 * r2 + k1 * r2 * r2 + k2 * r2 * r2 * r2;
    const float tan_ = p0 * yn1 + p1 * yn0;
    const float mlt = radial + 2.0f * tan_;
    const float yd0 = yn0 * mlt + p1 * r2;
    const float yd1 = yn1 * mlt + p0 * r2;
    float xy0 = f0 * yd0 + c0;
    float xy1 = f1 * yd1 + c1;
    const float width = 2.0f * cc0, height = 2.0f * cc1;

    const bool maskb = (xy0 >= 0.0f) && (xy1 >= 0.0f) && (xy0 < width) && (xy1 < height);
    if (maskb) {
      // clip(xy, -1, max(w,h)) is a no-op under mask, but keep it general:
      const float mx = fmaxf(width, height);
      xy0 = fminf(fmaxf(xy0, -1.0f), mx);
      xy1 = fminf(fmaxf(xy1, -1.0f), mx);
      const float h200 = 200.0f * sc0;
      xy0 = xy0 * (960.0f / h200) + 960.0f * (-cc0 / h200 + 0.5f);
      xy1 = xy1 * (512.0f / h200) + 512.0f * (-cc1 / h200 + 0.5f);
      xy0 *= (float)HM_W / 960.0f;
      xy1 *= (float)HM_H / 512.0f;
      float g0 = xy0 / (float)(HM_W - 1) * 2.0f - 1.0f;
      float g1 = xy1 / (float)(HM_H - 1) * 2.0f - 1.0f;
      g0 = fminf(fmaxf(g0, -1.1f), 1.1f);
      g1 = fminf(fmaxf(g1, -1.1f), 1.1f);
      const float px = (g0 + 1.0f) * 0.5f * (float)(HM_W - 1);
      const float py = (g1 + 1.0f) * 0.5f * (float)(HM_H - 1);

      const float x0f = floorf(px), y0f = floorf(py);
      const float dx = px - x0f, dy = py - y0f;
      const int x0 = (int)x0f, y0 = (int)y0f;
      float w00 = (1.0f - dx) * (1.0f - dy);
      float w10 = dx * (1.0f - dy);
      float w01 = (1.0f - dx) * dy;
      float w11 = dx * dy;
      const bool vx0 = (x0 >= 0) && (x0 < HM_W);
      const bool vx1 = (x0 + 1 >= 0) && (x0 + 1 < HM_W);
      const bool vy0 = (y0 >= 0) && (y0 < HM_H);
      const bool vy1 = (y0 + 1 >= 0) && (y0 + 1 < HM_H);
      w00 = (vx0 && vy0) ? w00 : 0.0f;
      w10 = (vx1 && vy0) ? w10 : 0.0f;
      w01 = (vx0 && vy1) ? w01 : 0.0f;
      w11 = (vx1 && vy1) ? w11 : 0.0f;
      const int ix0 = min(max(x0, 0), HM_W - 1);
      const int ix1 = min(max(x0 + 1, 0), HM_W - 1);
      const int iy0 = min(max(y0, 0), HM_H - 1);
      const int iy1 = min(max(y0 + 1, 0), HM_H - 1);
      const int o00 = iy0 * HM_W + ix0;
      const int o10 = iy0 * HM_W + ix1;
      const int o01 = iy1 * HM_W + ix0;
      const int o11 = iy1 * HM_W + ix1;

      const float* __restrict__ hb = heat + (size_t)bn * NJ * HM_H * HM_W;
#pragma unroll
      for (int j = 0; j < NJ; ++j) {
        const float* __restrict__ hj = hb + (size_t)j * (HM_H * HM_W);
        float s = hj[o00] * w00;
        s = fmaf(hj[o10], w10, s);
        s = fmaf(hj[o01], w01, s);
        s = fmaf(hj[o11], w11, s);
        num[j] += s;
      }
      den += 1.0f;
    }
  }

  const float invden = 1.0f / den;
  float* __restrict__ cb = cubes + (size_t)b * NJ * NVOX + v;
#pragma unroll
  for (int j = 0; j < NJ; ++j) {
    float val = num[j] * invden;
    val = fminf(fmaxf(val, 0.0f), 1.0f);   // nan_to_num + clip(0,1)
    cb[(size_t)j * NVOX] = val;
  }

  float* __restrict__ gr = grids + ((size_t)b * NVOX + v) * 3;
  gr[0] = gx; gr[1] = gy; gr[2] = gz;
}

extern "C" void kernel_launch(void* const* d_in, const int* in_sizes, int n_in,
                              void* d_out, int out_size, void* d_ws, size_t ws_size,
                              hipStream_t stream) {
  const float* heat    = (const float*)d_in[0];
  const float* gcenter = (const float*)d_in[1];
  const float* Rm      = (const float*)d_in[2];
  const float* Tm      = (const float*)d_in[3];
  const float* fo      = (const float*)d_in[4];
  const float* co      = (const float*)d_in[5];
  const float* ko      = (const float*)d_in[6];
  const float* po      = (const float*)d_in[7];
  const float* cco     = (const float*)d_in[8];
  const float* sco     = (const float*)d_in[9];

  float* cubes = (float*)d_out;                                  // [8,15,128000]
  float* grids = (float*)d_out + (size_t)NB * NJ * NVOX;         // [8,128000,3]

  dim3 grid(NVOX / 256, NB);   // 500 x 8 blocks, 256 threads (8 wave32s)
  project_layer_kernel<<<grid, 256, 0, stream>>>(
      heat, gcenter, Rm, Tm, fo, co, ko, po, cco, sco, cubes, grids);
}